// GCNModel_67611375174130
// MI455X (gfx1250) — compile-verified
//
#include <hip/hip_runtime.h>

#define N_FEATS 128

typedef __attribute__((ext_vector_type(2))) float v2f;
typedef __attribute__((ext_vector_type(8))) float v8f;

__device__ __forceinline__ void atomic_add_f32(float* p, float v) {
  // relaxed agent-scope fp32 add -> GLOBAL_ATOMIC_ADD_F32 (no-return form)
  __hip_atomic_fetch_add(p, v, __ATOMIC_RELAXED, __HIP_MEMORY_SCOPE_AGENT);
}

// ---------------- degree accumulation (1 thread per edge) ----------------
__global__ __launch_bounds__(256) void gcn_degrees(
    const int* __restrict__ src, const int* __restrict__ dst,
    float* __restrict__ deg_out, float* __restrict__ deg_in, int n_edges) {
  int i = blockIdx.x * blockDim.x + threadIdx.x;
  if (i >= n_edges) return;
  atomic_add_f32(&deg_out[src[i]], 1.0f);
  atomic_add_f32(&deg_in[dst[i]], 1.0f);
}

// ---------------- D^{-1/2} norms ----------------
__global__ __launch_bounds__(256) void gcn_norms(
    const float* __restrict__ deg_out, const float* __restrict__ deg_in,
    float* __restrict__ ns, float* __restrict__ nd, int n) {
  int i = blockIdx.x * blockDim.x + threadIdx.x;
  if (i >= n) return;
  ns[i] = rsqrtf(fmaxf(deg_out[i], 1.0f));
  nd[i] = rsqrtf(fmaxf(deg_in[i], 1.0f));
}

// ---------------- GEMM: out[N,128] = (A * norm_src[:,None]) @ W ----------------
// Block = 256 threads = 8 waves. Each block: 16 rows; wave w -> 16x16 tile at cols w*16.
// Exact fp32 via V_WMMA_F32_16X16X4_F32 (K stepped by 4, 32 WMMAs/wave).
__global__ __launch_bounds__(256) void gcn_gemm_wmma(
    const float* __restrict__ A, const float* __restrict__ norm_src,
    const float* __restrict__ W, float* __restrict__ out, int n_rows) {
  __shared__ float As[16][132];  // padded to dodge bank conflicts

  const int row0 = blockIdx.x << 4;

  // Cooperatively stage 16x128 A-tile, pre-scaled by norm_src (coalesced float4).
  for (int t = threadIdx.x; t < 512; t += 256) {
    const int r = t >> 5;          // 32 float4 per row
    const int c4 = (t & 31) << 2;
    const int row = row0 + r;
    if (row < n_rows) {
      const float4 v = *(const float4*)(A + (size_t)row * N_FEATS + c4);
      const float s = norm_src[row];
      As[r][c4 + 0] = v.x * s;
      As[r][c4 + 1] = v.y * s;
      As[r][c4 + 2] = v.z * s;
      As[r][c4 + 3] = v.w * s;
    } else {
      As[r][c4 + 0] = 0.f; As[r][c4 + 1] = 0.f;
      As[r][c4 + 2] = 0.f; As[r][c4 + 3] = 0.f;
    }
  }
  __syncthreads();

  const int wave = threadIdx.x >> 5;
  const int lane = threadIdx.x & 31;
  const int half = lane >> 4;   // lane half selects K pairs / M halves per ISA layout
  const int m    = lane & 15;
  const int col0 = wave << 4;

  v8f acc = {};
#pragma unroll
  for (int k = 0; k < N_FEATS; k += 4) {
    const int ka = k + (half << 1);  // half 0 -> k,k+1 ; half 1 -> k+2,k+3
    v2f a, b;
    // A 16x4 fragment: lane m holds row m, VGPR0=K even slot, VGPR1=next K
    a.x = As[m][ka];
    a.y = As[m][ka + 1];
    // B 4x16 fragment: lane m holds col (col0+m); rows striped by lane-half
    b.x = W[ka * N_FEATS + col0 + m];
    b.y = W[(ka + 1) * N_FEATS + col0 + m];
    acc = __builtin_amdgcn_wmma_f32_16x16x4_f32(
        /*neg_a=*/false, a, /*neg_b=*/false, b,
        /*c_mod=*/(short)0, acc, /*reuse_a=*/false, /*reuse_b=*/false);
  }

  // C/D layout: VGPR j -> M = j (lanes 0-15) or j+8 (lanes 16-31); N = m
#pragma unroll
  for (int j = 0; j < 8; ++j) {
    const int r = row0 + j + (half << 3);
    if (r < n_rows) out[(size_t)r * N_FEATS + col0 + m] = acc[j];
  }
}

// ---------------- edge aggregation: agg[dst] += h[src] ----------------
// One wave32 per edge: lane l moves floats [4l..4l+3] -> one full 512B row per wave.
__global__ __launch_bounds__(256) void gcn_agg(
    const float* __restrict__ h, const int* __restrict__ src,
    const int* __restrict__ dst, float* __restrict__ agg, int n_edges) {
  const int e    = (blockIdx.x * blockDim.x + threadIdx.x) >> 5;
  const int lane = threadIdx.x & 31;
  if (e >= n_edges) return;
  const int s = src[e];  // wave-uniform -> scalar load
  const int d = dst[e];
  const float4 v = *(const float4*)(h + (size_t)s * N_FEATS + (lane << 2));
  float* o = agg + (size_t)d * N_FEATS + (lane << 2);
  atomic_add_f32(o + 0, v.x);
  atomic_add_f32(o + 1, v.y);
  atomic_add_f32(o + 2, v.z);
  atomic_add_f32(o + 3, v.w);
}

// ---------------- finalize: out = [relu](agg * norm_dst[:,None] + b) ----------------
__global__ __launch_bounds__(256) void gcn_finalize(
    const float* __restrict__ agg, const float* __restrict__ norm_dst,
    const float* __restrict__ bias, float* __restrict__ out,
    int n_nodes, int do_relu) {
  const int i = blockIdx.x * blockDim.x + threadIdx.x;
  if (i >= n_nodes * 32) return;
  const int row = i >> 5;
  const int c4  = (i & 31) << 2;
  const float s = norm_dst[row];
  const float4 v = *(const float4*)(agg + (size_t)row * N_FEATS + c4);
  const float4 b = *(const float4*)(bias + c4);
  float4 r;
  r.x = v.x * s + b.x;
  r.y = v.y * s + b.y;
  r.z = v.z * s + b.z;
  r.w = v.w * s + b.w;
  if (do_relu) {
    r.x = fmaxf(r.x, 0.f); r.y = fmaxf(r.y, 0.f);
    r.z = fmaxf(r.z, 0.f); r.w = fmaxf(r.w, 0.f);
  }
  *(float4*)(out + (size_t)row * N_FEATS + c4) = r;
}

static inline size_t align256(size_t x) { return (x + 255) & ~(size_t)255; }

extern "C" void kernel_launch(void* const* d_in, const int* in_sizes, int n_in,
                              void* d_out, int out_size, void* d_ws, size_t ws_size,
                              hipStream_t stream) {
  (void)n_in; (void)out_size; (void)ws_size;
  const float* feats = (const float*)d_in[0];
  const int*   src   = (const int*)d_in[1];
  const int*   dst   = (const int*)d_in[2];
  const float* W1    = (const float*)d_in[3];
  const float* b1    = (const float*)d_in[4];
  const float* W2    = (const float*)d_in[5];
  const float* b2    = (const float*)d_in[6];
  float* out = (float*)d_out;

  const int n_nodes = in_sizes[0] / N_FEATS;
  const int n_edges = in_sizes[1];

  const size_t nb = align256((size_t)n_nodes * sizeof(float));
  const size_t hb = align256((size_t)n_nodes * N_FEATS * sizeof(float));
  char* p = (char*)d_ws;
  float* deg_out = (float*)p; p += nb;
  float* deg_in  = (float*)p; p += nb;
  float* ns      = (float*)p; p += nb;
  float* nd      = (float*)p; p += nb;
  float* bufA    = (float*)p; p += hb;   // GEMM out / layer activations
  float* bufB    = (float*)p; p += hb;   // aggregation accumulator / GEMM2 out

  const int tb = 256;
  const size_t node_bytes = (size_t)n_nodes * sizeof(float);
  const size_t feat_bytes = (size_t)n_nodes * N_FEATS * sizeof(float);

  // degrees + norms
  hipMemsetAsync(deg_out, 0, node_bytes, stream);
  hipMemsetAsync(deg_in,  0, node_bytes, stream);
  gcn_degrees<<<(n_edges + tb - 1) / tb, tb, 0, stream>>>(src, dst, deg_out, deg_in, n_edges);
  gcn_norms<<<(n_nodes + tb - 1) / tb, tb, 0, stream>>>(deg_out, deg_in, ns, nd, n_nodes);

  const int gemm_grid = (n_nodes + 15) / 16;
  const int agg_grid  = (n_edges + 7) / 8;          // 8 edges (waves) per block
  const int fin_grid  = (n_nodes * 32 + tb - 1) / tb;

  // ---- layer 1 ----
  gcn_gemm_wmma<<<gemm_grid, 256, 0, stream>>>(feats, ns, W1, bufA, n_nodes);
  hipMemsetAsync(bufB, 0, feat_bytes, stream);
  gcn_agg<<<agg_grid, 256, 0, stream>>>(bufA, src, dst, bufB, n_edges);
  gcn_finalize<<<fin_grid, tb, 0, stream>>>(bufB, nd, b1, bufA, n_nodes, /*relu=*/1);

  // ---- layer 2 ----
  gcn_gemm_wmma<<<gemm_grid, 256, 0, stream>>>(bufA, ns, W2, bufB, n_nodes);
  hipMemsetAsync(out, 0, feat_bytes, stream);
  gcn_agg<<<agg_grid, 256, 0, stream>>>(bufB, src, dst, out, n_edges);
  gcn_finalize<<<fin_grid, tb, 0, stream>>>(out, nd, b2, out, n_nodes, /*relu=*/0);
}